// DePOI_57827439673447
// MI455X (gfx1250) — compile-verified
//
#include <hip/hip_runtime.h>
#include <hip/hip_bf16.h>
#include <math.h>

// Problem constants (from reference)
#define BB 4
#define LL 200
#define HH 256
#define NHD 4
#define HSZ 64
#define NEGV (-4294967295.0f)   // -(2^32)+1

typedef __attribute__((ext_vector_type(2))) float v2f;
typedef __attribute__((ext_vector_type(4))) float v4f;
typedef __attribute__((ext_vector_type(8))) float v8f;

// ---------------------------------------------------------------------------
// Stage 1: projection GEMM  out[m,n] = sum_k X[m,k] * W[n,k] + bias[n] (+pos)
// One wave (32 lanes) computes one 16x16 tile via 64 x V_WMMA_F32_16X16X4_F32.
// Fp32 WMMA keeps exact fp32 semantics vs the fp32 reference.
// Templated on HAS_POS so the epilogue is branch-free.
// ---------------------------------------------------------------------------
template <bool HAS_POS>
__global__ __launch_bounds__(32)
void proj16(const float* __restrict__ X, const float* __restrict__ W,
            const float* __restrict__ bias, const float* __restrict__ pos,
            float* __restrict__ out)
{
    const int lane    = threadIdx.x;          // 0..31, EXEC all-ones
    const int rowBase = blockIdx.x * 16;      // 50 tiles over 800 rows
    const int colBase = blockIdx.y * 16;      // 16 tiles over 256 cols
    const int l16     = lane & 15;
    const int koff    = (lane >> 4) * 2;      // lanes 0-15: K+{0,1}; 16-31: K+{2,3}

    const float* arow = X + (size_t)(rowBase + l16) * HH;  // A[m, :]
    const float* brow = W + (size_t)(colBase + l16) * HH;  // B[k,n] = W[n,k]

    v8f c = {};
    #pragma unroll 4
    for (int k0 = 0; k0 < HH; k0 += 4) {
        v2f a  = *(const v2f*)(arow + k0 + koff);   // global_load_b64, 8B aligned
        v2f bf = *(const v2f*)(brow + k0 + koff);
        c = __builtin_amdgcn_wmma_f32_16x16x4_f32(
                /*neg_a=*/false, a, /*neg_b=*/false, bf,
                /*c_mod=*/(short)0, c, /*reuse_a=*/false, /*reuse_b=*/false);
    }

    // Epilogue: C/D layout -> lane%16 = N, VGPR r -> M = r + 8*(lane/16)
    const int n     = colBase + l16;
    const float bv  = bias[n];
    const int mBase = rowBase + ((lane >> 4) << 3);
    #pragma unroll
    for (int r = 0; r < 8; ++r) {
        const int m = mBase + r;
        float v = c[r] + bv;
        if (HAS_POS) v += pos[(size_t)m * HH + n];  // fold abs_pos_K / abs_pos_V
        out[(size_t)m * HH + n] = v;
    }
}

// ---------------------------------------------------------------------------
// Stage 2+3 fused: per (b,l) block -> causal scores, softmax, output.
// Only m <= l rows of the big (B,L,L,H) tensors are ever touched (causal),
// halving HBM traffic vs the reference. Those streams are read-once ->
// non-temporal loads (TH=NT) so the reused Q/Kp/Vp stay L2-resident.
// ---------------------------------------------------------------------------
__global__ __launch_bounds__(256)
void attn_fused(const float* __restrict__ Q,  const float* __restrict__ Kp,
                const float* __restrict__ Vp, const float* __restrict__ tK,
                const float* __restrict__ dK, const float* __restrict__ tV,
                const float* __restrict__ dV, const unsigned char* __restrict__ tmask,
                float* __restrict__ out)
{
    const int l    = blockIdx.x;
    const int b    = blockIdx.y;
    const int tid  = threadIdx.x;       // 0..255
    const int wave = tid >> 5;          // 8 waves
    const int lane = tid & 31;
    const int M    = l + 1;             // valid keys (causal)

    __shared__ __align__(16) float w_sm[LL * NHD];   // scores/probs [m][h]

    const size_t qrow   = (size_t)(b * LL + l) * HH;
    const size_t rowOff = (size_t)(b * LL + l) * LL * HH;  // (b,l) slab of tK/dK/tV/dV

    // Per-lane Q fragments: d = half*128 + lane*4  (head = half*2 + lane/16)
    const v4f q0 = *(const v4f*)(Q + qrow + 0   + lane * 4);
    const v4f q1 = *(const v4f*)(Q + qrow + 128 + lane * 4);

    // ---- scores: each wave owns rows m = wave, wave+8, ... ----
    for (int m = wave; m < M; m += 8) {
        const float* tKm = tK + rowOff + (size_t)m * HH;
        const float* dKm = dK + rowOff + (size_t)m * HH;
        const float* kpm = Kp + (size_t)(b * LL + m) * HH;
        const bool tm = tmask[b * LL + m] != 0;

        if (m + 8 < M) {  // stream this wave's next row toward the caches
            __builtin_prefetch(tKm + 8 * HH + lane * 8, 0, 0);  // 32 lanes x 32B = 1KB row
            __builtin_prefetch(dKm + 8 * HH + lane * 8, 0, 0);
        }
        #pragma unroll
        for (int half = 0; half < 2; ++half) {
            const int off = half * 128 + lane * 4;
            const v4f t4 = __builtin_nontemporal_load((const v4f*)(tKm + off));
            const v4f d4 = __builtin_nontemporal_load((const v4f*)(dKm + off));
            const v4f k4 = *(const v4f*)(kpm + off);       // K + abs_pos_K (L2 reuse)
            const v4f qv = half ? q1 : q0;
            float s = qv.x * (t4.x + d4.x + k4.x) + qv.y * (t4.y + d4.y + k4.y)
                    + qv.z * (t4.z + d4.z + k4.z) + qv.w * (t4.w + d4.w + k4.w);
            // reduce 16 lanes (one head = 64 floats = 16 lanes x float4)
            s += __shfl_xor(s, 1, 32);
            s += __shfl_xor(s, 2, 32);
            s += __shfl_xor(s, 4, 32);
            s += __shfl_xor(s, 8, 32);
            if ((lane & 15) == 0) {
                const int head = half * 2 + (lane >> 4);
                w_sm[m * NHD + head] = tm ? NEGV : s * 0.125f;  // 1/sqrt(64)
            }
        }
    }
    __syncthreads();

    // ---- softmax over m for each head: wave h handles head h ----
    if (wave < NHD) {
        const int h = wave;
        float mx = -3.402823466e38f;
        for (int m = lane; m < M; m += 32) mx = fmaxf(mx, w_sm[m * NHD + h]);
        #pragma unroll
        for (int i = 16; i >= 1; i >>= 1) mx = fmaxf(mx, __shfl_xor(mx, i, 32));
        float sum = 0.0f;
        for (int m = lane; m < M; m += 32) {
            const float e = __expf(w_sm[m * NHD + h] - mx);
            w_sm[m * NHD + h] = e;
            sum += e;
        }
        #pragma unroll
        for (int i = 16; i >= 1; i >>= 1) sum += __shfl_xor(sum, i, 32);
        const float inv = 1.0f / sum;
        for (int m = lane; m < M; m += 32) w_sm[m * NHD + h] *= inv;
    }
    __syncthreads();

    // ---- output: thread tid owns column tid (head = tid/64), fully coalesced ----
    const int col = tid;
    const int h   = col >> 6;
    const float* tVr = tV + rowOff + col;
    const float* dVr = dV + rowOff + col;
    const float* vpr = Vp + (size_t)(b * LL) * HH + col;   // V + abs_pos_V (L2 reuse)
    float acc = 0.0f;
    for (int m = 0; m < M; ++m) {
        if (m + 8 < M) {
            __builtin_prefetch(tVr + (size_t)(m + 8) * HH, 0, 0);
            __builtin_prefetch(dVr + (size_t)(m + 8) * HH, 0, 0);
        }
        const float a = w_sm[m * NHD + h];                 // LDS broadcast per 64 thr
        const float v = vpr[(size_t)m * HH]
                      + __builtin_nontemporal_load(tVr + (size_t)m * HH)
                      + __builtin_nontemporal_load(dVr + (size_t)m * HH);
        acc = fmaf(a, v, acc);
    }
    out[qrow + col] = acc;
}

// ---------------------------------------------------------------------------
extern "C" void kernel_launch(void* const* d_in, const int* in_sizes, int n_in,
                              void* d_out, int out_size, void* d_ws, size_t ws_size,
                              hipStream_t stream) {
    const float* queries = (const float*)d_in[0];
    const float* keys    = (const float*)d_in[1];
    const float* tK      = (const float*)d_in[2];
    const float* tV      = (const float*)d_in[3];
    const float* dK      = (const float*)d_in[4];
    const float* dV      = (const float*)d_in[5];
    const float* posK    = (const float*)d_in[6];
    const float* posV    = (const float*)d_in[7];
    const float* Qw      = (const float*)d_in[8];
    const float* Qb      = (const float*)d_in[9];
    const float* Kw      = (const float*)d_in[10];
    const float* Kb      = (const float*)d_in[11];
    const float* Vw      = (const float*)d_in[12];
    const float* Vb      = (const float*)d_in[13];
    const unsigned char* tmask = (const unsigned char*)d_in[14];
    // d_in[15] attn_mask: known strict-causal -> applied structurally (m <= l)

    const int ROWS = BB * LL;                 // 800
    float* Qp = (float*)d_ws;                 // [800,256]
    float* Kp = Qp + (size_t)ROWS * HH;       // K proj + abs_pos_K
    float* Vp = Kp + (size_t)ROWS * HH;       // V proj + abs_pos_V

    dim3 g1(ROWS / 16, HH / 16), b1(32);
    proj16<false><<<g1, b1, 0, stream>>>(queries, Qw, Qb, nullptr, Qp);
    proj16<true ><<<g1, b1, 0, stream>>>(keys,    Kw, Kb, posK,    Kp);
    proj16<true ><<<g1, b1, 0, stream>>>(keys,    Vw, Vb, posV,    Vp);

    dim3 g2(LL, BB), b2(256);
    attn_fused<<<g2, b2, 0, stream>>>(Qp, Kp, Vp, tK, dK, tV, dV, tmask,
                                      (float*)d_out);
}